// DynamicRouter_56324201119926
// MI455X (gfx1250) — compile-verified
//
#include <hip/hip_runtime.h>

typedef __attribute__((ext_vector_type(2))) float v2f;
typedef __attribute__((ext_vector_type(8))) float v8f;

constexpr int D = 768;                 // model dim
constexpr int E = 8;                   // experts
constexpr int TOKENS_PER_WAVE  = 16;   // one 16x16 WMMA tile per wave
constexpr int WAVES_PER_BLOCK  = 8;    // 256 threads
constexpr int TOKENS_PER_BLOCK = TOKENS_PER_WAVE * WAVES_PER_BLOCK; // 128
constexpr int WSTRIDE = D + 2;         // 770; 770 % 64 == 2 -> conflict-free B reads
constexpr float NOISE_STD = 0.1f;

__global__ __launch_bounds__(256)
void moe_router_wmma_f32(const float* __restrict__ x,
                         const float* __restrict__ W,
                         const float* __restrict__ bias,
                         const float* __restrict__ noise,
                         float* __restrict__ out,
                         float* __restrict__ idxOut)
{
    // W tile padded to 16 "experts" (rows 8..15 zero) for the 16-wide N dim.
    __shared__ float ldsW[16 * WSTRIDE];                    // 49280 B
    __shared__ float ldsC[WAVES_PER_BLOCK * 16 * 16];       //  8192 B

    const int tid = threadIdx.x;

    // Zero the whole padded W region (ghost rows + pad lanes), then copy W.
    for (int i = tid; i < 16 * WSTRIDE; i += 256) ldsW[i] = 0.0f;
    __syncthreads();
    for (int i = tid; i < E * D; i += 256) {
        int e = i / D;
        int k = i - e * D;
        ldsW[e * WSTRIDE + k] = W[i];
    }
    __syncthreads();

    const int lane = tid & 31;
    const int wave = tid >> 5;
    const int m    = lane & 15;   // token row (A) / expert col (B)
    const int kh   = lane >> 4;   // K half: lanes 0-15 hold K{0,1}, 16-31 hold K{2,3}
    const int tokBase = blockIdx.x * TOKENS_PER_BLOCK + wave * TOKENS_PER_WAVE;

    // A fragment source: row of x for token (tokBase+m), offset by K-half.
    const float* xrow = x + (size_t)(tokBase + m) * D + 2 * kh;
    // B fragment source: LDS row of expert m (zero for m>=8), offset by K-half.
    const float* wrow = &ldsW[m * WSTRIDE + 2 * kh];

    // Two accumulators to break the WMMA->WMMA accumulation chain.
    v8f c0 = {};
    v8f c1 = {};
#pragma unroll 4
    for (int k = 0; k < D; k += 8) {
        v2f a0 = *(const v2f*)(xrow + k);
        v2f b0 = *(const v2f*)(wrow + k);
        c0 = __builtin_amdgcn_wmma_f32_16x16x4_f32(false, a0, false, b0,
                                                   (short)0, c0, false, false);
        v2f a1 = *(const v2f*)(xrow + k + 4);
        v2f b1 = *(const v2f*)(wrow + k + 4);
        c1 = __builtin_amdgcn_wmma_f32_16x16x4_f32(false, a1, false, b1,
                                                   (short)0, c1, false, false);
    }
    v8f c = c0 + c1;

    // C layout: VGPR v, lanes 0-15 -> row v, lanes 16-31 -> row v+8, col = m.
    float* myTile = &ldsC[wave * 256];
#pragma unroll
    for (int v = 0; v < 8; ++v)
        myTile[(v + 8 * kh) * 16 + m] = c[v];
    __syncthreads();

    // One lane per token finishes: bias + noise, top-2, sparse softmax.
    if (lane < 16) {
        const int tok = tokBase + lane;
        float lg[E];
#pragma unroll
        for (int e = 0; e < E; ++e)
            lg[e] = myTile[lane * 16 + e] + bias[e]
                  + NOISE_STD * noise[(size_t)tok * E + e];

        // top-1 (strict > keeps lowest index on ties, matching lax.top_k)
        int i1 = 0;
        float m1 = lg[0];
#pragma unroll
        for (int e = 1; e < E; ++e)
            if (lg[e] > m1) { m1 = lg[e]; i1 = e; }
        // top-2 among e != i1
        int i2 = (i1 == 0) ? 1 : 0;
        float m2 = lg[i2];
#pragma unroll
        for (int e = 0; e < E; ++e)
            if (e != i1 && lg[e] > m2) { m2 = lg[e]; i2 = e; }

        // softmax over {m1, m2}; all other experts get exp(-inf) = 0.
        const float t  = __expf(m2 - m1);      // <= 1, numerically safe
        const float rs = 1.0f / (1.0f + t);
        const float p1 = rs;
        const float p2 = t * rs;

#pragma unroll
        for (int e = 0; e < E; ++e) {
            float p = (e == i1) ? p1 : ((e == i2) ? p2 : 0.0f);
            out[(size_t)tok * E + e] = p;
        }
        idxOut[(size_t)tok * 2 + 0] = (float)i1;
        idxOut[(size_t)tok * 2 + 1] = (float)i2;
    }
}

extern "C" void kernel_launch(void* const* d_in, const int* in_sizes, int n_in,
                              void* d_out, int out_size, void* d_ws, size_t ws_size,
                              hipStream_t stream) {
    const float* x     = (const float*)d_in[0];   // [B,S,D]
    const float* W     = (const float*)d_in[1];   // [E,D]
    const float* b     = (const float*)d_in[2];   // [E]
    const float* noise = (const float*)d_in[3];   // [B,S,E]

    const int tokens = in_sizes[3] / E;           // B*S = 32768
    float* out    = (float*)d_out;                // [tokens, E] probs
    float* idxOut = out + (size_t)tokens * E;     // [tokens, 2] indices (as f32)

    const int blocks = tokens / TOKENS_PER_BLOCK; // exact for this shape
    hipLaunchKernelGGL(moe_router_wmma_f32, dim3(blocks), dim3(256), 0, stream,
                       x, W, b, noise, out, idxOut);
}